// get_model_45810121179160
// MI455X (gfx1250) — compile-verified
//
#include <hip/hip_runtime.h>

#define B_  4
#define L_  2048
#define DM_ 256
#define DI_ 512
#define DS_ 16
#define DR_ 16
#define DC_ 4

typedef __attribute__((ext_vector_type(2))) float v2f;
typedef __attribute__((ext_vector_type(8))) float v8f;

// D = A(16x4) * B(4x16) + C, full fp32 matrix path on CDNA5.
__device__ __forceinline__ v8f wmma4(v2f a, v2f b, v8f c) {
  return __builtin_amdgcn_wmma_f32_16x16x4_f32(
      /*neg_a=*/false, a, /*neg_b=*/false, b,
      /*c_mod=*/(short)0, c, /*reuse_a=*/false, /*reuse_b=*/false);
}

__device__ __forceinline__ float silu_(float x) { return x / (1.0f + __expf(-x)); }
__device__ __forceinline__ float softplus_(float x) {
  return (x > 20.0f) ? x : log1pf(__expf(x));
}

// ---------------------------------------------------------------------------
// hidden_states (B, L, DM) -> hsT (B, DM, L)
// ---------------------------------------------------------------------------
__global__ void transpose_hs(const float* __restrict__ hs, float* __restrict__ hsT) {
  int idx = blockIdx.x * blockDim.x + threadIdx.x;      // over B*L*DM, exact
  int dm = idx % DM_;
  int l  = (idx / DM_) % L_;
  int b  = idx / (DM_ * L_);
  hsT[((size_t)b * DM_ + dm) * L_ + l] = hs[idx];
}

// ---------------------------------------------------------------------------
// "chan-rows" GEMM:  out[b][m][l] = sum_k W[m][k] * act[b][k][l]  (+epilogue)
//   W   : (Mtot, KDIM) row-major           (A matrix, contiguous per lane)
//   act : (B, actC, L) channel-major       (B matrix, lanes coalesced in l)
//   out : (B, Mtot, L)                     (coalesced half-wave stores)
// One wave computes a 16(M) x 64(N=l) tile. EPI==1: softplus(x + bias[m]).
// ---------------------------------------------------------------------------
template <int KDIM, int EPI>
__global__ void gemm_chanrows(const float* __restrict__ W,
                              const float* __restrict__ act, int actC,
                              float* __restrict__ out, int Mtot,
                              const float* __restrict__ bias) {
  const int lane = threadIdx.x & 31;
  const int h = lane >> 4;        // K half
  const int j = lane & 15;        // A row m / B col n
  const int l0 = blockIdx.x * 64;
  const int m0 = blockIdx.y * 16;
  const int b  = blockIdx.z;

  const float* Wr = W + (size_t)(m0 + j) * KDIM;
  const float* Ab = act + (size_t)b * actC * L_;

  v8f acc0 = {}, acc1 = {}, acc2 = {}, acc3 = {};
#pragma unroll 4
  for (int k0 = 0; k0 < KDIM; k0 += 4) {
    v2f a;
    a.x = Wr[k0 + 2 * h];
    a.y = Wr[k0 + 2 * h + 1];
    const float* r = Ab + (size_t)(k0 + 2 * h) * L_ + l0 + j;
    v2f b0, b1, b2, b3;
    b0.x = r[0];  b0.y = r[L_];
    b1.x = r[16]; b1.y = r[L_ + 16];
    b2.x = r[32]; b2.y = r[L_ + 32];
    b3.x = r[48]; b3.y = r[L_ + 48];
    acc0 = wmma4(a, b0, acc0);
    acc1 = wmma4(a, b1, acc1);
    acc2 = wmma4(a, b2, acc2);
    acc3 = wmma4(a, b3, acc3);
  }

  float* Ob = out + (size_t)b * Mtot * L_;
#pragma unroll
  for (int v = 0; v < 8; ++v) {
    int mr = m0 + v + 8 * h;
    float e0 = acc0[v], e1 = acc1[v], e2 = acc2[v], e3 = acc3[v];
    if (EPI == 1) {
      float bi = bias[mr];
      e0 = softplus_(e0 + bi); e1 = softplus_(e1 + bi);
      e2 = softplus_(e2 + bi); e3 = softplus_(e3 + bi);
    }
    size_t base = (size_t)mr * L_ + l0 + j;
    Ob[base] = e0; Ob[base + 16] = e1; Ob[base + 32] = e2; Ob[base + 48] = e3;
  }
}

// ---------------------------------------------------------------------------
// Causal depthwise conv (k=4) + SiLU over xz[:, :DI, :] in logical (maybe
// reversed) sequence coordinates.  xout (B, DI, L) logical coords.
// ---------------------------------------------------------------------------
__global__ void conv_silu(const float* __restrict__ xz,
                          const float* __restrict__ cw,
                          const float* __restrict__ cb,
                          float* __restrict__ xout, int rev) {
  int t = blockIdx.x * blockDim.x + threadIdx.x;   // logical l
  int d = blockIdx.y;
  int b = blockIdx.z;
  const float* s = xz + ((size_t)b * (2 * DI_) + d) * L_;
  float acc = cb[d];
#pragma unroll
  for (int i = 0; i < DC_; ++i) {
    int tt = t - (DC_ - 1) + i;
    float v = 0.0f;
    if (tt >= 0) v = s[rev ? (L_ - 1 - tt) : tt];
    acc += cw[d * DC_ + i] * v;
  }
  xout[((size_t)b * DI_ + d) * L_ + t] = silu_(acc);
}

// ---------------------------------------------------------------------------
// Selective scan. One lane per (d-half, n): lane = 16*dh + n.
// Wave covers 2 channels x 16 states; h kept in a register; Sum_n h*C via
// 16-lane shfl_xor tree; lane n==0 applies D-skip + SiLU(z) gate and stores.
// ---------------------------------------------------------------------------
__global__ void scan_kernel(const float* __restrict__ xz,     // z gate source
                            const float* __restrict__ u,      // (B, DI, L) logical
                            const float* __restrict__ xdbl,   // (B, 48, L): dt|B|C
                            const float* __restrict__ delta,  // (B, DI, L)
                            const float* __restrict__ A_log,  // (DI, DS)
                            const float* __restrict__ Dp,     // (DI)
                            float* __restrict__ y,            // (B, DI, L) original coords
                            int rev, int accumulate) {
  const int lane = threadIdx.x & 31;
  const int n  = lane & 15;
  const int dh = lane >> 4;
  const int wave = threadIdx.x >> 5;
  const int d = blockIdx.x * 16 + wave * 2 + dh;
  const int b = blockIdx.y;

  const float And = -__expf(A_log[(size_t)d * DS_ + n]);
  const float Dd  = Dp[d];

  const float* dl = delta + ((size_t)b * DI_ + d) * L_;
  const float* ur = u     + ((size_t)b * DI_ + d) * L_;
  const float* Bm = xdbl  + ((size_t)b * 48 + DR_ + n) * L_;
  const float* Cm = xdbl  + ((size_t)b * 48 + DR_ + DS_ + n) * L_;
  const float* zr = xz    + ((size_t)b * (2 * DI_) + DI_ + d) * L_;
  float* yr = y + ((size_t)b * DI_ + d) * L_;

  float h = 0.0f;
  for (int t = 0; t < L_; ++t) {
    float de = dl[t];
    float uu = ur[t];
    h = __expf(de * And) * h + (de * uu) * Bm[t];
    float p = h * Cm[t];
    p += __shfl_xor(p, 8, 16);
    p += __shfl_xor(p, 4, 16);
    p += __shfl_xor(p, 2, 16);
    p += __shfl_xor(p, 1, 16);
    if (n == 0) {
      int tm = rev ? (L_ - 1 - t) : t;
      float z  = zr[tm];
      float yv = (p + uu * Dd) * silu_(z);
      if (accumulate) yv += yr[tm];
      yr[tm] = yv;
    }
  }
}

// ---------------------------------------------------------------------------
// out_proj GEMM: out[b][l][o] = sum_d y[b][d][l] * Wo[o][d]
// rows = (b,l) tiles of 16, cols = o tiles of 64; final stores coalesced.
// ---------------------------------------------------------------------------
__global__ void gemm_out(const float* __restrict__ y,   // (B, DI, L)
                         const float* __restrict__ Wo,  // (DM, DI) row-major
                         float* __restrict__ out) {     // (B, L, DM)
  const int lane = threadIdx.x & 31;
  const int h = lane >> 4;
  const int j = lane & 15;
  const int o0 = blockIdx.x * 64;
  const int l0 = blockIdx.y * 16;
  const int b  = blockIdx.z;

  const float* Yb = y + (size_t)b * DI_ * L_;
  const float* Wr = Wo + (size_t)(o0 + j) * DI_;

  v8f acc0 = {}, acc1 = {}, acc2 = {}, acc3 = {};
#pragma unroll 4
  for (int k0 = 0; k0 < DI_; k0 += 4) {
    const float* ac = Yb + (size_t)(k0 + 2 * h) * L_ + l0 + j;
    v2f a;
    a.x = ac[0];
    a.y = ac[L_];
    const float* wr = Wr + k0 + 2 * h;
    v2f b0, b1, b2, b3;
    b0.x = wr[0];            b0.y = wr[1];
    b1.x = wr[16 * DI_];     b1.y = wr[16 * DI_ + 1];
    b2.x = wr[32 * DI_];     b2.y = wr[32 * DI_ + 1];
    b3.x = wr[48 * DI_];     b3.y = wr[48 * DI_ + 1];
    acc0 = wmma4(a, b0, acc0);
    acc1 = wmma4(a, b1, acc1);
    acc2 = wmma4(a, b2, acc2);
    acc3 = wmma4(a, b3, acc3);
  }
#pragma unroll
  for (int v = 0; v < 8; ++v) {
    size_t row = (size_t)b * L_ + l0 + v + 8 * h;
    float* orow = out + row * DM_ + o0 + j;
    orow[0] = acc0[v]; orow[16] = acc1[v]; orow[32] = acc2[v]; orow[48] = acc3[v];
  }
}

// ---------------------------------------------------------------------------
extern "C" void kernel_launch(void* const* d_in, const int* in_sizes, int n_in,
                              void* d_out, int out_size, void* d_ws, size_t ws_size,
                              hipStream_t stream) {
  (void)in_sizes; (void)n_in; (void)out_size; (void)ws_size;

  const float* hs      = (const float*)d_in[0];
  const float* ipw     = (const float*)d_in[1];
  const float* cw[2]   = {(const float*)d_in[2],  (const float*)d_in[9]};
  const float* cb[2]   = {(const float*)d_in[3],  (const float*)d_in[10]};
  const float* xpw[2]  = {(const float*)d_in[4],  (const float*)d_in[11]};
  const float* dtw[2]  = {(const float*)d_in[5],  (const float*)d_in[12]};
  const float* dtb[2]  = {(const float*)d_in[6],  (const float*)d_in[13]};
  const float* Alog[2] = {(const float*)d_in[7],  (const float*)d_in[14]};
  const float* Dp[2]   = {(const float*)d_in[8],  (const float*)d_in[15]};
  const float* opw     = (const float*)d_in[16];
  float* out = (float*)d_out;

  float* w = (float*)d_ws;
  float* hsT   = w; w += (size_t)B_ * DM_ * L_;            // 2,097,152 f
  float* xz    = w; w += (size_t)B_ * 2 * DI_ * L_;        // 8,388,608 f
  float* xbuf  = w; w += (size_t)B_ * DI_ * L_;            // 4,194,304 f
  float* xdbl  = w; w += (size_t)B_ * (DR_ + 2 * DS_) * L_;//   393,216 f
  float* delta = w; w += (size_t)B_ * DI_ * L_;            // 4,194,304 f
  float* ybuf  = w; w += (size_t)B_ * DI_ * L_;            // 4,194,304 f

  // 1) transpose hidden_states to (B, DM, L)
  transpose_hs<<<dim3((B_ * L_ * DM_) / 256), 256, 0, stream>>>(hs, hsT);

  // 2) in_proj: xz (B, 2*DI, L)
  gemm_chanrows<DM_, 0><<<dim3(L_ / 64, (2 * DI_) / 16, B_), 32, 0, stream>>>(
      ipw, hsT, DM_, xz, 2 * DI_, nullptr);

  // 3) forward + backward Mamba directions (backward = reversed indexing,
  //    scan accumulates into ybuf)
  for (int dir = 0; dir < 2; ++dir) {
    conv_silu<<<dim3(L_ / 256, DI_, B_), 256, 0, stream>>>(
        xz, cw[dir], cb[dir], xbuf, dir);
    gemm_chanrows<DI_, 0><<<dim3(L_ / 64, 48 / 16, B_), 32, 0, stream>>>(
        xpw[dir], xbuf, DI_, xdbl, 48, nullptr);
    gemm_chanrows<DR_, 1><<<dim3(L_ / 64, DI_ / 16, B_), 32, 0, stream>>>(
        dtw[dir], xdbl, 48, delta, DI_, dtb[dir]);
    scan_kernel<<<dim3(DI_ / 16, B_), 256, 0, stream>>>(
        xz, xbuf, xdbl, delta, Alog[dir], Dp[dir], ybuf, dir, dir);
  }

  // 4) out_proj: out (B, L, DM)
  gemm_out<<<dim3(DM_ / 64, L_ / 16, B_), 32, 0, stream>>>(ybuf, opw, out);
}